// FWPBlock_69965017252760
// MI455X (gfx1250) — compile-verified
//
#include <hip/hip_runtime.h>
#include <math.h>

#define NB 4
#define NT 512
#define NF 512
#define ND 256

typedef __attribute__((ext_vector_type(2))) float v2f;
typedef __attribute__((ext_vector_type(4))) float v4f;
typedef __attribute__((ext_vector_type(8))) float v8f;

__device__ __forceinline__ float wave_reduce_add(float x) {
    x += __shfl_xor(x, 16, 32);
    x += __shfl_xor(x, 8, 32);
    x += __shfl_xor(x, 4, 32);
    x += __shfl_xor(x, 2, 32);
    x += __shfl_xor(x, 1, 32);
    return x;
}

__device__ __forceinline__ float dot8(v4f a, v4f b, v4f c, v4f d) {
    return a.x * b.x + a.y * b.y + a.z * b.z + a.w * b.w +
           c.x * d.x + c.y * d.y + c.z * d.z + c.w * d.w;
}

// ---------------- LayerNorm: one wave per (b,t) row of F=512 ----------------
__global__ void ln_kernel(const float* __restrict__ x,
                          const float* __restrict__ gamma,
                          const float* __restrict__ beta,
                          float* __restrict__ xn) {
    const int wave = threadIdx.x >> 5, lane = threadIdx.x & 31;
    const int row = blockIdx.x * 8 + wave;            // 2048 rows
    const float* xr = x + (size_t)row * NF;
    float vals[16];
    float s = 0.f, ss = 0.f;
#pragma unroll
    for (int u = 0; u < 16; ++u) {
        float v = xr[lane + 32 * u];
        vals[u] = v; s += v; ss += v * v;
    }
#pragma unroll
    for (int m = 16; m; m >>= 1) {
        s  += __shfl_xor(s,  m, 32);
        ss += __shfl_xor(ss, m, 32);
    }
    const float mu  = s * (1.f / NF);
    const float var = ss * (1.f / NF) - mu * mu;
    const float r   = rsqrtf(var + 1e-5f);
    float* xo = xn + (size_t)row * NF;
#pragma unroll
    for (int u = 0; u < 16; ++u) {
        int j = lane + 32 * u;
        xo[j] = (vals[u] - mu) * r * gamma[j] + beta[j];
    }
}

// ------- WMMA fp32 GEMM: O = xn @ W   (M=2048, K=512, N=256), z picks W -----
__global__ void gemm_kernel(const float* __restrict__ xn,
                            const float* __restrict__ Wk, const float* __restrict__ Wq,
                            const float* __restrict__ Wv, const float* __restrict__ Wsc,
                            const float* __restrict__ bsc,
                            float* __restrict__ Kb, float* __restrict__ Qb,
                            float* __restrict__ Vb, float* __restrict__ SCb) {
    const int lane = threadIdx.x;                 // block = 1 wave32
    const int m0 = blockIdx.x * 16;
    const int n0 = blockIdx.y * 16;
    const int z  = blockIdx.z;
    const float* W; float* O;
    switch (z) {
        case 0:  W = Wk;  O = Kb;  break;
        case 1:  W = Wq;  O = Qb;  break;
        case 2:  W = Wv;  O = Vb;  break;
        default: W = Wsc; O = SCb; break;
    }
    const int half = lane >> 4;                   // K-half select (ISA A/B layout)
    const int l16  = lane & 15;
    const int col  = n0 + l16;
    const float* arow = xn + (size_t)(m0 + l16) * NF;
    v8f acc = {};
    for (int k0 = 0; k0 < NF; k0 += 4) {
        const int kk = k0 + 2 * half;
        v2f a, b;
        a.x = arow[kk];
        a.y = arow[kk + 1];
        b.x = W[(size_t)kk * ND + col];
        b.y = W[(size_t)(kk + 1) * ND + col];
        // D = A(16x4,f32) * B(4x16,f32) + C
        acc = __builtin_amdgcn_wmma_f32_16x16x4_f32(
            false, a, false, b, (short)0, acc, false, false);
    }
    const float bias = (z == 3) ? bsc[col] : 0.f;  // single uniform branch
#pragma unroll
    for (int r = 0; r < 8; ++r) {
        const int row = m0 + r + 8 * half;        // C/D layout: VGPR r -> M = r + 8*(lane/16)
        O[(size_t)row * ND + col] = acc[r] + bias;
    }
}

// --- beta = sigmoid(xn@Wbeta); relu + sum-normalize K and Q; wave per row ---
__global__ void act_kernel(const float* __restrict__ xn,
                           const float* __restrict__ Wbeta,
                           float* __restrict__ Kb, float* __restrict__ Qb,
                           float* __restrict__ betab) {
    const int wave = threadIdx.x >> 5, lane = threadIdx.x & 31;
    const int row = blockIdx.x * 8 + wave;        // 2048 rows
    const float* xr = xn + (size_t)row * NF;
    float zacc = 0.f;
#pragma unroll
    for (int u = 0; u < 16; ++u) {
        int j = lane + 32 * u;
        zacc += xr[j] * Wbeta[j];
    }
    zacc = wave_reduce_add(zacc);
    if (lane == 0) betab[row] = 1.f / (1.f + expf(-zacc));

    float* kr = Kb + (size_t)row * ND;
    v4f klo = ((const v4f*)kr)[lane];
    v4f khi = ((const v4f*)(kr + 128))[lane];
#pragma unroll
    for (int c = 0; c < 4; ++c) { klo[c] = fmaxf(klo[c], 0.f); khi[c] = fmaxf(khi[c], 0.f); }
    float ks = wave_reduce_add(klo.x + klo.y + klo.z + klo.w + khi.x + khi.y + khi.z + khi.w);
    float kn = 1.f / (1e-5f + ks);
    ((v4f*)kr)[lane] = klo * kn;
    ((v4f*)(kr + 128))[lane] = khi * kn;

    float* qr = Qb + (size_t)row * ND;
    v4f qlo = ((const v4f*)qr)[lane];
    v4f qhi = ((const v4f*)(qr + 128))[lane];
#pragma unroll
    for (int c = 0; c < 4; ++c) { qlo[c] = fmaxf(qlo[c], 0.f); qhi[c] = fmaxf(qhi[c], 0.f); }
    float qs = wave_reduce_add(qlo.x + qlo.y + qlo.z + qlo.w + qhi.x + qhi.y + qhi.z + qhi.w);
    float qn = 1.f / (1e-5f + qs);
    ((v4f*)qr)[lane] = qlo * qn;
    ((v4f*)(qr + 128))[lane] = qhi * qn;
}

// --- delta-rule scan: one wave owns one state row i of one batch b ----------
// s_i(t) = s_i(t-1) + beta_t * (v_i(t) - <s_i, k_t>) * k_t
// states streamed out with non-temporal b128 stores (512MB, don't pollute L2).
// k/q rows for t+1 prefetched (global_prefetch_b8) to hide per-step latency.
__global__ void scan_kernel(const float* __restrict__ Kb, const float* __restrict__ Qb,
                            const float* __restrict__ Vb, const float* __restrict__ betab,
                            const float* __restrict__ SCb, const float* __restrict__ state0,
                            float* __restrict__ y, float* __restrict__ states) {
    const int wave = threadIdx.x >> 5, lane = threadIdx.x & 31;
    const int b = blockIdx.x >> 5;                     // 32 blocks per batch
    const int i = (blockIdx.x & 31) * 8 + wave;        // state row 0..255

    const float* s0 = state0 + ((size_t)b * ND + i) * ND;
    v4f slo = ((const v4f*)s0)[lane];                  // j = 4*lane .. 4*lane+3
    v4f shi = ((const v4f*)(s0 + 128))[lane];          // j = 128+4*lane ..

    for (int t = 0; t < NT; ++t) {
        const size_t rt = (size_t)b * NT + t;

        if (t + 1 < NT) {                              // prefetch next step's k/q row
            const float* kn = Kb + (rt + 1) * ND;
            const float* qn = Qb + (rt + 1) * ND;
            __builtin_prefetch(kn + lane * 8, 0, 3);   // 32 lanes x 32B = full 1KB row
            __builtin_prefetch(qn + lane * 8, 0, 3);
        }

        const float* kp = Kb + rt * ND;
        const v4f klo = ((const v4f*)kp)[lane];
        const v4f khi = ((const v4f*)(kp + 128))[lane];

        float d = wave_reduce_add(dot8(slo, klo, shi, khi));   // vbar_i
        const float coef = betab[rt] * (Vb[rt * ND + i] - d);  // beta*(v - vbar)
        slo += coef * klo;
        shi += coef * khi;

        float* sp = states + ((size_t)rt * ND + i) * ND;
        __builtin_nontemporal_store(slo, (v4f*)sp + lane);
        __builtin_nontemporal_store(shi, (v4f*)(sp + 128) + lane);

        const float* qp = Qb + rt * ND;
        const v4f qlo = ((const v4f*)qp)[lane];
        const v4f qhi = ((const v4f*)(qp + 128))[lane];
        float yd = wave_reduce_add(dot8(slo, qlo, shi, qhi));
        if (lane == 0) y[rt * ND + i] = yd + SCb[rt * ND + i];
    }
}

extern "C" void kernel_launch(void* const* d_in, const int* in_sizes, int n_in,
                              void* d_out, int out_size, void* d_ws, size_t ws_size,
                              hipStream_t stream) {
    const float* x       = (const float*)d_in[0];
    const float* state0  = (const float*)d_in[1];
    const float* Wk      = (const float*)d_in[2];
    const float* Wq      = (const float*)d_in[3];
    const float* Wv      = (const float*)d_in[4];
    const float* Wbeta   = (const float*)d_in[5];
    const float* gamma   = (const float*)d_in[6];
    const float* beta_ln = (const float*)d_in[7];
    const float* Wsc     = (const float*)d_in[8];
    const float* bsc     = (const float*)d_in[9];

    float* y      = (float*)d_out;                          // [B,T,D]
    float* states = y + (size_t)NB * NT * ND;               // [B,T,D,D]

    float* ws    = (float*)d_ws;
    float* xn    = ws;                                      // 2048*512
    float* Kb    = xn + (size_t)NB * NT * NF;               // 2048*256
    float* Qb    = Kb + (size_t)NB * NT * ND;
    float* Vb    = Qb + (size_t)NB * NT * ND;
    float* SCb   = Vb + (size_t)NB * NT * ND;
    float* betab = SCb + (size_t)NB * NT * ND;              // 2048

    ln_kernel<<<256, 256, 0, stream>>>(x, gamma, beta_ln, xn);
    gemm_kernel<<<dim3(NB * NT / 16, ND / 16, 4), 32, 0, stream>>>(
        xn, Wk, Wq, Wv, Wsc, bsc, Kb, Qb, Vb, SCb);
    act_kernel<<<256, 256, 0, stream>>>(xn, Wbeta, Kb, Qb, betab);
    scan_kernel<<<NB * (ND / 8), 256, 0, stream>>>(
        Kb, Qb, Vb, betab, SCb, state0, y, states);
}